// MAE_CreateDecoderInput_Raw_35751307772079
// MI455X (gfx1250) — compile-verified
//
#include <hip/hip_runtime.h>
#include <cstdint>

// MAE decoder-input assembly for MI455X (gfx1250), single fused launch.
// Memory-bound (~600 MB traffic, AI ~0.45 flop/B -> ~26 us at 23.3 TB/s HBM).
// CDNA5 paths: async global<->LDS DMA (ASYNCcnt) for the mask-row scatter,
// b128 vector memory + non-temporal hints, wave32 shuffle reductions for LN.

#define MAE_B   128
#define MAE_T   1024
#define MAE_K   512
#define MAE_NM  768
#define MAE_NU  256
#define MAE_LN_EPS 1e-5f

#define MAE_NBLK_A ((MAE_B * MAE_NM) / 4)   // 24576 copy blocks (4 rows each)
#define MAE_NBLK_B (MAE_B * MAE_NU)         // 32768 LN blocks (1 row each)

typedef float v4f __attribute__((ext_vector_type(4)));

// ---------------------------------------------------------------------------
// Fused kernel.
//   blockIdx.x <  MAE_NBLK_A : scatter mask_embedding rows via async LDS DMA
//                              (one wave32 per 512-float row, 4x b128/lane,
//                               data never touches VGPRs)
//   blockIdx.x >= MAE_NBLK_A : x = LayerNorm(enc+pos)*gamma+beta scattered to
//                              dec (one 128-thread block per row)
// Branch is block-uniform; the two paths write disjoint output rows, so the
// output needs no zero-init (mask_id | unmask_id partition [0,T) per sample).
// ---------------------------------------------------------------------------
__global__ __launch_bounds__(128) void mae_decoder_input_fused_kernel(
    const float* __restrict__ enc,        // (B, NU, K)
    const float* __restrict__ pos,        // (B, NU, K)
    const float* __restrict__ mask_emb,   // (B, NM, K)
    const float* __restrict__ gamma,      // (K)
    const float* __restrict__ beta,       // (K)
    const int*   __restrict__ mask_id,    // (B, NM)
    const int*   __restrict__ unmask_id,  // (B, NU)
    float*       __restrict__ dec)        // (B, T, K)
{
  __shared__ float lds[4][MAE_K];  // 8 KB: DMA staging (path A) / scratch (path B)

  const int tid  = threadIdx.x;
  const int lane = tid & 31;
  const int wib  = tid >> 5;

  if (blockIdx.x < MAE_NBLK_A) {
    // ---------------- Path A: mask_embedding row scatter via async DMA ------
    const long long row = (long long)blockIdx.x * 4 + wib;  // (b, i) flattened
    const int b = (int)(row / MAE_NM);
    const int t = mask_id[row];                             // destination token
    const float* src = mask_emb + row * MAE_K;
    float*       dst = dec + ((long long)b * MAE_T + t) * MAE_K;

    // Low 32 bits of the generic shared-pointer = LDS byte offset.
    const unsigned lbase = (unsigned)(uintptr_t)(&lds[wib][0]);

#pragma unroll
    for (int k = 0; k < 4; ++k) {
      const unsigned           la = lbase + (unsigned)(k * 512 + lane * 16);
      const unsigned long long ga =
          (unsigned long long)(uintptr_t)(src + k * 128 + lane * 4);
      asm volatile("global_load_async_to_lds_b128 %0, %1, off"
                   :: "v"(la), "v"(ga) : "memory");
    }
    // All async loads landed in LDS (ASYNCcnt==0) before the stores read it.
    asm volatile("s_wait_asynccnt 0x0" ::: "memory");
#pragma unroll
    for (int k = 0; k < 4; ++k) {
      const unsigned           la = lbase + (unsigned)(k * 512 + lane * 16);
      const unsigned long long ga =
          (unsigned long long)(uintptr_t)(dst + k * 128 + lane * 4);
      asm volatile("global_store_async_from_lds_b128 %0, %1, off"
                   :: "v"(ga), "v"(la) : "memory");
    }
    // S_ENDPGM's implicit wait-idle drains ASYNCcnt before the wave exits.
  } else {
    // ---------------- Path B: LayerNorm(enc+pos)*gamma+beta, scattered ------
    const long long row = (long long)blockIdx.x - MAE_NBLK_A;  // (b, j) flat
    const int b = (int)(row / MAE_NU);

    // Read-once streams: non-temporal b128 loads.
    v4f v = __builtin_nontemporal_load((const v4f*)(enc + row * MAE_K) + tid);
    v4f q = __builtin_nontemporal_load((const v4f*)(pos + row * MAE_K) + tid);
    v += q;

    float s  = v.x + v.y + v.z + v.w;
    float ss = fmaf(v.x, v.x, fmaf(v.y, v.y, fmaf(v.z, v.z, v.w * v.w)));
#pragma unroll
    for (int off = 16; off > 0; off >>= 1) {
      s  += __shfl_xor(s,  off, 32);
      ss += __shfl_xor(ss, off, 32);
    }

    float* red = &lds[0][0];               // reuse staging LDS: 8 floats
    if (lane == 0) { red[wib] = s; red[4 + wib] = ss; }
    __syncthreads();
    s  = red[0] + red[1] + red[2] + red[3];
    ss = red[4] + red[5] + red[6] + red[7];

    const float inv = 1.0f / (float)MAE_K;
    const float mu  = s * inv;
    const float var = ss * inv - mu * mu;
    const float rs  = 1.0f / sqrtf(var + MAE_LN_EPS);

    const v4f g  = *((const v4f*)gamma + tid);
    const v4f be = *((const v4f*)beta  + tid);
    const v4f y  = (v - mu) * rs * g + be;

    const int t = unmask_id[row];
    float* dst = dec + ((long long)b * MAE_T + t) * MAE_K;
    __builtin_nontemporal_store(y, (v4f*)dst + tid);  // write-once stream
  }
}

// ---------------------------------------------------------------------------
extern "C" void kernel_launch(void* const* d_in, const int* in_sizes, int n_in,
                              void* d_out, int out_size, void* d_ws, size_t ws_size,
                              hipStream_t stream) {
  (void)in_sizes; (void)n_in; (void)out_size; (void)d_ws; (void)ws_size;
  const float* enc       = (const float*)d_in[0];   // encoder_output
  const float* pos       = (const float*)d_in[1];   // unmasked_positions
  const float* mask_emb  = (const float*)d_in[2];   // mask_embedding
  const float* gamma     = (const float*)d_in[3];
  const float* beta      = (const float*)d_in[4];
  const int*   mask_id   = (const int*)d_in[5];
  const int*   unmask_id = (const int*)d_in[6];
  float*       dec       = (float*)d_out;

  mae_decoder_input_fused_kernel<<<dim3(MAE_NBLK_A + MAE_NBLK_B), dim3(128), 0,
                                   stream>>>(
      enc, pos, mask_emb, gamma, beta, mask_id, unmask_id, dec);
}